// FixedHiddenMLP_53712861003867
// MI455X (gfx1250) — compile-verified
//
#include <hip/hip_runtime.h>
#include <hip/hip_bf16.h>
#include <math.h>

typedef __attribute__((ext_vector_type(2))) float v2f;
typedef __attribute__((ext_vector_type(8))) float v8f;

#define NB   512          // blocks in main kernel (partials count)
#define TPB  256          // 8 waves per block (wave32)
#define WPB  (TPB / 32)

// relu in a single VALU op: v_med3_f32(x, 0, +inf) == clamp(x, 0, +inf)
__device__ __forceinline__ float relu1(float x) {
    return __builtin_amdgcn_fmed3f(x, 0.0f, __builtin_inff());
}

// ---------------------------------------------------------------------------
// Kernel 0: fuse the two linear layers on device.
//   M[k][j] = sum_t W[t][k] * R[t][j]   (X @ W^T @ R == X @ M), padded to 32 cols
//   c[j]    = sum_t b[t]    * R[t][j] + 1, padded cols = 0  (so relu(0+0)=0)
// ---------------------------------------------------------------------------
__global__ void fuse_prep_kernel(const float* __restrict__ W,
                                 const float* __restrict__ b,
                                 const float* __restrict__ R,
                                 float* __restrict__ Mpad,   // [20][32]
                                 float* __restrict__ cpad) { // [32]
    int j = threadIdx.x;           // 0..31 (one column each)
    if (j >= 32) return;
    float cj = 0.0f;
    if (j < 20) {
        for (int t = 0; t < 20; ++t) cj += b[t] * R[t * 20 + j];
        cj += 1.0f;
    }
    cpad[j] = cj;
    for (int k = 0; k < 20; ++k) {
        float m = 0.0f;
        if (j < 20)
            for (int t = 0; t < 20; ++t) m += W[t * 20 + k] * R[t * 20 + j];
        Mpad[k * 32 + j] = m;
    }
}

// ---------------------------------------------------------------------------
// Kernel 1: streaming GEMM + relu + reduction, one 16-row tile per wave/iter.
//   D(16x16) = A(16x4,f32) x B(4x16,f32) + C via v_wmma_f32_16x16x4_f32.
//   K=20 -> 5 chunks; N=20 -> two 16-col tiles (cols 20..31 zero-padded).
//   Bias pre-seeded in the WMMA accumulator; relu via v_med3; scalar loop ctrl.
// ---------------------------------------------------------------------------
__global__ void __launch_bounds__(TPB)
mlp_wmma_sum_kernel(const float* __restrict__ X,
                    const float* __restrict__ Mpad,
                    const float* __restrict__ cpad,
                    float* __restrict__ partials,
                    int numTiles) {
    const int lane = threadIdx.x & 31;
    const int wib  = threadIdx.x >> 5;
    // wave-uniform values forced into SGPRs -> SALU loop control, EXEC untouched
    const int waveId   = __builtin_amdgcn_readfirstlane(blockIdx.x * WPB + wib);
    const int numWaves = gridDim.x * WPB;

    const int ln = lane & 15;        // A: row m, B: col n, bias index
    const int kk = (lane >> 4) * 2;  // K sub-offset per half-wave (ISA A/B layout)

    // B registers: 5 K-chunks x 2 N-tiles, identical in every wave (L2-resident).
    v2f B0[5], B1[5];
#pragma unroll
    for (int q = 0; q < 5; ++q) {
        const int kr = 4 * q + kk;
        B0[q].x = Mpad[kr * 32 + ln];
        B0[q].y = Mpad[(kr + 1) * 32 + ln];
        B1[q].x = Mpad[kr * 32 + 16 + ln];
        B1[q].y = Mpad[(kr + 1) * 32 + 16 + ln];
    }
    const float cb0 = cpad[ln];
    const float cb1 = cpad[16 + ln];

    v8f acc0 = {};                   // running sum(relu(.)) in vector form
    v8f acc1 = {};
    const float* baseA = X + (size_t)ln * 20 + kk;

    for (int tile = waveId; tile < numTiles; tile += numWaves) {   // SALU loop
        const float* p = baseA + (size_t)tile * (16 * 20);
        v8f c0, c1;                  // accumulators pre-seeded with bias
#pragma unroll
        for (int t = 0; t < 8; ++t) { c0[t] = cb0; c1[t] = cb1; }
#pragma unroll
        for (int q = 0; q < 5; ++q) {
            v2f a = *(const v2f*)(p + 4 * q);   // 8B-aligned (even float offset)
            c0 = __builtin_amdgcn_wmma_f32_16x16x4_f32(
                     false, a, false, B0[q], (short)0, c0, false, false);
            c1 = __builtin_amdgcn_wmma_f32_16x16x4_f32(
                     false, a, false, B1[q], (short)0, c1, false, false);
        }
#pragma unroll
        for (int t = 0; t < 8; ++t) {
            acc0[t] += relu1(c0[t]);            // padded cols give relu(0)=0
            acc1[t] += relu1(c1[t]);
        }
    }

    float lsum = 0.0f;
#pragma unroll
    for (int t = 0; t < 8; ++t) lsum += acc0[t] + acc1[t];

    // wave32 tree reduction
#pragma unroll
    for (int off = 16; off > 0; off >>= 1)
        lsum += __shfl_xor(lsum, off, 32);

    __shared__ float wsum[WPB];
    if (lane == 0) wsum[wib] = lsum;
    __syncthreads();
    if (threadIdx.x == 0) {
        float s = 0.0f;
        for (int w = 0; w < WPB; ++w) s += wsum[w];
        partials[blockIdx.x] = s;
    }
}

// ---------------------------------------------------------------------------
// Kernel 2: deterministic finalize — tail rows (scalar), sum partials,
// count halvings of sum(|h|), write total * 2^-n.
// ---------------------------------------------------------------------------
__global__ void finalize_kernel(const float* __restrict__ partials, int nb,
                                const float* __restrict__ X,
                                const float* __restrict__ Mpad,
                                const float* __restrict__ cpad,
                                int tailStart, int rows,
                                float* __restrict__ out) {
    __shared__ float sh[256];
    float s = 0.0f;
    for (int i = threadIdx.x; i < nb; i += blockDim.x) s += partials[i];

    // tail rows not covered by 16-row tiles (none for 2M rows, kept for safety)
    const int tailElems = (rows - tailStart) * 20;
    for (int idx = threadIdx.x; idx < tailElems; idx += blockDim.x) {
        const int r = tailStart + idx / 20;
        const int j = idx % 20;
        float v = cpad[j];
        for (int k = 0; k < 20; ++k) v += X[(size_t)r * 20 + k] * Mpad[k * 32 + j];
        s += relu1(v);
    }

    sh[threadIdx.x] = s;
    __syncthreads();
    for (int stride = 128; stride > 0; stride >>= 1) {
        if (threadIdx.x < stride) sh[threadIdx.x] += sh[threadIdx.x + stride];
        __syncthreads();
    }
    if (threadIdx.x == 0) {
        const float total = sh[0];
        float s0 = total;
        int n = 0;
        while (s0 > 1.0f && n < 4096) { s0 *= 0.5f; ++n; }
        out[0] = ldexpf(total, -n);   // exact power-of-two scaling
    }
}

// ---------------------------------------------------------------------------
extern "C" void kernel_launch(void* const* d_in, const int* in_sizes, int n_in,
                              void* d_out, int out_size, void* d_ws, size_t ws_size,
                              hipStream_t stream) {
    const float* X = (const float*)d_in[0];
    const float* W = (const float*)d_in[1];
    const float* b = (const float*)d_in[2];
    const float* R = (const float*)d_in[3];
    float* out = (float*)d_out;

    float* ws       = (float*)d_ws;
    float* Mpad     = ws;         // 20*32 = 640 floats
    float* cpad     = ws + 640;   // 32 floats
    float* partials = ws + 672;   // NB floats

    const int rows      = in_sizes[0] / 20;   // 2,000,000
    const int numTiles  = rows / 16;          // 125,000 (exact)
    const int tailStart = numTiles * 16;

    fuse_prep_kernel<<<1, 32, 0, stream>>>(W, b, R, Mpad, cpad);
    mlp_wmma_sum_kernel<<<NB, TPB, 0, stream>>>(X, Mpad, cpad, partials, numTiles);
    finalize_kernel<<<1, 256, 0, stream>>>(partials, NB, X, Mpad, cpad,
                                           tailStart, rows, out);
}